// RegionLoss_67568425501004
// MI455X (gfx1250) — compile-verified
//
#include <hip/hip_runtime.h>
#include <hip/hip_bf16.h>
#include <stdint.h>

// Problem constants (match reference)
#define NB 32
#define NT 50
#define NA 5
#define NC 3
#define NH 128
#define NW 128
#define ATTRS 10                 // 7 + NC
#define PLANE (NH * NW)          // 16384
#define TOTAL (NB * NA * PLANE)  // 2,621,440 cells
#define NTGT (NB * NT)           // 1600 targets
#define IGNORE_THRES 0.5f
#define OBJ_SCALE 1.0f
#define NOOBJ_SCALE 100.0f
#define EPSF 1e-16f

// Launch shape
#define NBLK 512
#define NTHR 256

typedef __attribute__((ext_vector_type(2))) float v2f;
typedef __attribute__((ext_vector_type(8))) float v8f;

__device__ __forceinline__ float sigmoidf_(float x) { return 1.0f / (1.0f + expf(-x)); }
__device__ __forceinline__ float clampp_(float p) {
    return fminf(fmaxf(p, 1e-12f), 1.0f - 1e-12f);
}

// ---------------------------------------------------------------------------
// Kernel 0: zero the scatter grids (workspace is poisoned, not re-poisoned).
// ---------------------------------------------------------------------------
__global__ void zero_ws_kernel(int* __restrict__ objGrid, int* __restrict__ ignGridW) {
    int idx = blockIdx.x * blockDim.x + threadIdx.x;
    int stride = gridDim.x * blockDim.x;
    for (int k = idx; k < TOTAL; k += stride) objGrid[k] = 0;
    for (int k = idx; k < TOTAL / 4; k += stride) ignGridW[k] = 0;  // u8 grid, 4 at a time
}

// ---------------------------------------------------------------------------
// Kernel 1: build targets (1600 scatter threads).
//   objGrid[cell] = (target idx + 1) via atomicMax  (deterministic winner)
//   ignGrid[cell] = 1 where anchor IoU > 0.5        (all writers write 1)
//   records[idx]  = {tx, ty, tw, th, tim, tre, label, 0}
// ---------------------------------------------------------------------------
__global__ void build_targets_kernel(const float* __restrict__ targets,
                                     const float* __restrict__ anchors,
                                     int* __restrict__ objGrid,
                                     uint8_t* __restrict__ ignGrid,
                                     float* __restrict__ records) {
    int idx = blockIdx.x * blockDim.x + threadIdx.x;
    if (idx >= NTGT) return;
    int b = idx / NT;
    const float* t = targets + (size_t)idx * 7;
    float lab = t[0], x = t[1], y = t[2], w = t[3], l = t[4], im = t[5], re = t[6];
    float asum = fabsf(lab) + fabsf(x) + fabsf(y) + fabsf(w) + fabsf(l) + fabsf(im) + fabsf(re);
    if (!(asum > 0.0f)) return;  // invalid target -> dropped

    float gx = x * NW, gy = y * NH, gw = w * NW, gl = l * NH;
    int gi = min(max((int)gx, 0), NW - 1);
    int gj = min(max((int)gy, 0), NH - 1);

    int best = 0;
    float bestIou = -1.0f, bestAw = 1.0f, bestAl = 1.0f;
#pragma unroll
    for (int a = 0; a < NA; ++a) {
        float aw = anchors[a * 4 + 0];
        float al = anchors[a * 4 + 1];
        float inter = fminf(gw, aw) * fminf(gl, al);
        float iou = inter / (gw * gl + aw * al - inter + EPSF);
        if (iou > bestIou) { bestIou = iou; best = a; bestAw = aw; bestAl = al; }
        if (iou > IGNORE_THRES)
            ignGrid[((b * NA + a) * NH + gj) * NW + gi] = (uint8_t)1;
    }

    float* r = records + (size_t)idx * 8;
    r[0] = gx - (float)gi;
    r[1] = gy - (float)gj;
    r[2] = logf(gw / bestAw + EPSF);
    r[3] = logf(gl / bestAl + EPSF);
    r[4] = im;
    r[5] = re;
    r[6] = lab;
    r[7] = 0.0f;

    atomicMax(&objGrid[((b * NA + best) * NH + gj) * NW + gi], idx + 1);
}

// ---------------------------------------------------------------------------
// Kernel 2: main masked reduction over all 2.62M cells.
// Only the conf plane is read for every cell; the other 9 planes only at the
// ~1600 obj cells. 11 accumulators per thread -> wave shfl -> LDS -> one
// 16-float partial record per block.
// ---------------------------------------------------------------------------
__global__ void region_loss_main_kernel(const float* __restrict__ out,
                                        const int* __restrict__ objGrid,
                                        const uint8_t* __restrict__ ignGrid,
                                        const float* __restrict__ records,
                                        float* __restrict__ partials) {
    float acc[11];
#pragma unroll
    for (int k = 0; k < 11; ++k) acc[k] = 0.0f;

    int tid = threadIdx.x;
    for (int cell = blockIdx.x * NTHR + tid; cell < TOTAL; cell += NBLK * NTHR) {
        int ba = cell >> 14;                       // b*NA + a   (PLANE == 1<<14)
        int ji = cell & (PLANE - 1);               // j*NW + i
        const float* base = out + (size_t)ba * (ATTRS * PLANE) + ji;

        float pconf = sigmoidf_(base[6 * PLANE]);  // always needed
        int rec = objGrid[cell];
        bool ign = ignGrid[cell] != 0;

        if (rec > 0) {  // obj cell (rare)
            acc[9] += 1.0f;  // sum(mo)
            const float* r = records + (size_t)(rec - 1) * 8;
            float px = sigmoidf_(base[0 * PLANE]);
            float py = sigmoidf_(base[1 * PLANE]);
            float dx = px - r[0], dy = py - r[1];
            float dw = base[2 * PLANE] - r[2];
            float dh = base[3 * PLANE] - r[3];
            float di = base[4 * PLANE] - r[4];
            float dr = base[5 * PLANE] - r[5];
            acc[0] += dx * dx; acc[1] += dy * dy;
            acc[2] += dw * dw; acc[3] += dh * dh;
            acc[4] += di * di; acc[5] += dr * dr;
            acc[6] += -logf(clampp_(pconf));       // BCE, t = 1
            int lab = (int)r[6];
#pragma unroll
            for (int c = 0; c < NC; ++c) {
                float pc = clampp_(sigmoidf_(base[(7 + c) * PLANE]));
                acc[8] += (c == lab) ? -logf(pc) : -log1pf(-pc);
            }
        } else if (!ign) {  // noobj cell
            acc[10] += 1.0f;  // sum(mn)
            acc[7] += -log1pf(-clampp_(pconf));    // BCE, t = 0
        }
    }

    // intra-block reduction: wave32 shuffle, then across the 8 waves via LDS
    __shared__ float red[11 * 8];
    int lane = tid & 31, wave = tid >> 5;
#pragma unroll
    for (int k = 0; k < 11; ++k) {
        float v = acc[k];
        for (int off = 16; off > 0; off >>= 1) v += __shfl_down(v, off, 32);
        if (lane == 0) red[k * 8 + wave] = v;
    }
    __syncthreads();
    if (tid < 16) {
        float s = 0.0f;
        if (tid < 11) {
#pragma unroll
            for (int k = 0; k < 8; ++k) s += red[tid * 8 + k];
        }
        partials[blockIdx.x * 16 + tid] = s;  // slots 11..15 written as 0
    }
}

// ---------------------------------------------------------------------------
// Kernel 3: final cross-block reduction as a ones-matrix multiply on the
// CDNA5 matrix pipe: V_WMMA_F32_16X16X4_F32, D = A(16x4) * ones(4x16) + C.
// A layout (ISA 7.12.2, 32-bit A 16x4): lanes 0-15 hold row=lane K=0..1 in
// VGPR0..1; lanes 16-31 hold row=lane-16 K=2..3. Each WMMA folds 4 block
// partial records into the f32 accumulator; D rows replicate across N.
// ---------------------------------------------------------------------------
__global__ void final_wmma_reduce_kernel(const float* __restrict__ partials,
                                         float* __restrict__ out) {
    int lane = threadIdx.x;  // 32 threads, all active (EXEC all-1s for WMMA)
    v8f c = {0.f, 0.f, 0.f, 0.f, 0.f, 0.f, 0.f, 0.f};
    v2f ones; ones.x = 1.0f; ones.y = 1.0f;
    int half = (lane >= 16) ? 2 : 0;  // which K-pair this half-wave supplies
    int row = lane & 15;

    for (int t = 0; t < NBLK / 4; ++t) {
        int base = (4 * t + half) * 16 + row;
        v2f a;
        a.x = partials[base];        // K = half+0
        a.y = partials[base + 16];   // K = half+1
        // (neg_a, A, neg_b, B, c_mod, C, reuse_a, reuse_b)
        c = __builtin_amdgcn_wmma_f32_16x16x4_f32(false, a, false, ones,
                                                  (short)0, c, false, false);
    }

    // D layout: lanes 0-15 VGPR v -> row v; lanes 16-31 VGPR v -> row v+8.
    __shared__ float sums[16];
    if (lane == 0) {
#pragma unroll
        for (int v = 0; v < 8; ++v) sums[v] = c[v];
    }
    if (lane == 16) {
#pragma unroll
        for (int v = 0; v < 8; ++v) sums[8 + v] = c[v];
    }
    __syncthreads();
    if (lane == 0) {
        float smo = sums[9], smn = sums[10];
        float inv_mo = 1.0f / fmaxf(smo, 1.0f);
        float loss_box = (sums[0] + sums[1] + sums[2] + sums[3] + sums[4] + sums[5]) * inv_mo;
        float loss_conf = OBJ_SCALE * sums[6] * inv_mo +
                          NOOBJ_SCALE * sums[7] / fmaxf(smn, 1.0f);
        float loss_cls = sums[8] / fmaxf(smo * (float)NC, 1.0f);
        out[0] = loss_box + loss_conf + loss_cls;
    }
}

// ---------------------------------------------------------------------------
extern "C" void kernel_launch(void* const* d_in, const int* in_sizes, int n_in,
                              void* d_out, int out_size, void* d_ws, size_t ws_size,
                              hipStream_t stream) {
    const float* output  = (const float*)d_in[0];  // (NB, NA*ATTRS, NH, NW)
    const float* targets = (const float*)d_in[1];  // (NB, NT, 7)
    const float* anchors = (const float*)d_in[2];  // (NA, 4)

    char* ws = (char*)d_ws;
    int*      objGrid  = (int*)ws;                               // TOTAL int32
    uint8_t*  ignGrid  = (uint8_t*)(ws + (size_t)TOTAL * 4);     // TOTAL u8
    float*    records  = (float*)(ws + (size_t)TOTAL * 5);       // NTGT * 8 f32
    float*    partials = (float*)(ws + (size_t)TOTAL * 5 + (size_t)NTGT * 8 * 4); // NBLK*16 f32

    zero_ws_kernel<<<NBLK, NTHR, 0, stream>>>(objGrid, (int*)ignGrid);
    build_targets_kernel<<<(NTGT + NTHR - 1) / NTHR, NTHR, 0, stream>>>(
        targets, anchors, objGrid, ignGrid, records);
    region_loss_main_kernel<<<NBLK, NTHR, 0, stream>>>(
        output, objGrid, ignGrid, records, partials);
    final_wmma_reduce_kernel<<<1, 32, 0, stream>>>(partials, (float*)d_out);
}